// GENConvModule_88364657148498
// MI455X (gfx1250) — compile-verified
//
#include <hip/hip_runtime.h>
#include <hip/hip_bf16.h>
#include <math.h>

typedef float v2f __attribute__((ext_vector_type(2)));
typedef float v8f __attribute__((ext_vector_type(8)));

#define NNODES 100000
#define NEDGES 1000000
#define DFEAT  64
#define D2     128
#define GEPS   1e-7f

// ---------------------------------------------------------------- utilities
__global__ void k_zero(float* __restrict__ p, long n) {
    long i = (long)blockIdx.x * blockDim.x + threadIdx.x;
    long stride = (long)gridDim.x * blockDim.x;
    for (; i < n; i += stride) p[i] = 0.0f;
}

// ------------------------------------------------- edge pass 1: segment max
// msg = relu(x[src]) + eps  (> 0), so int compare == float compare.
__global__ void k_edge_max(const float* __restrict__ x, const int* __restrict__ src,
                           const int* __restrict__ dst, float* __restrict__ mx) {
    int tid = blockIdx.x * blockDim.x + threadIdx.x;
    if (tid >= NEDGES * 16) return;
    int e  = tid >> 4;
    int d4 = (tid & 15) * 4;
    int s = src[e], t = dst[e];
    float4 xv = *(const float4*)(x + (long)s * DFEAT + d4);
    float m0 = fmaxf(xv.x, 0.f) + GEPS;
    float m1 = fmaxf(xv.y, 0.f) + GEPS;
    float m2 = fmaxf(xv.z, 0.f) + GEPS;
    float m3 = fmaxf(xv.w, 0.f) + GEPS;
    int* p = (int*)(mx + (long)t * DFEAT + d4);
    atomicMax(p + 0, __float_as_int(m0));
    atomicMax(p + 1, __float_as_int(m1));
    atomicMax(p + 2, __float_as_int(m2));
    atomicMax(p + 3, __float_as_int(m3));
}

// ------------------------------------------ edge pass 2: sum of exp(msg-mx)
__global__ void k_edge_expsum(const float* __restrict__ x, const int* __restrict__ src,
                              const int* __restrict__ dst, const float* __restrict__ mx,
                              float* __restrict__ denom) {
    int tid = blockIdx.x * blockDim.x + threadIdx.x;
    if (tid >= NEDGES * 16) return;
    int e  = tid >> 4;
    int d4 = (tid & 15) * 4;
    int s = src[e], t = dst[e];
    float4 xv = *(const float4*)(x + (long)s * DFEAT + d4);
    float4 mv = *(const float4*)(mx + (long)t * DFEAT + d4);
    float e0 = __expf(fmaxf(xv.x, 0.f) + GEPS - mv.x);
    float e1 = __expf(fmaxf(xv.y, 0.f) + GEPS - mv.y);
    float e2 = __expf(fmaxf(xv.z, 0.f) + GEPS - mv.z);
    float e3 = __expf(fmaxf(xv.w, 0.f) + GEPS - mv.w);
    float* p = denom + (long)t * DFEAT + d4;
    atomicAdd(p + 0, e0);
    atomicAdd(p + 1, e1);
    atomicAdd(p + 2, e2);
    atomicAdd(p + 3, e3);
}

// ------------------------- edge pass 3: agg += msg * exp(msg-mx)/(denom+eps)
__global__ void k_edge_agg(const float* __restrict__ x, const int* __restrict__ src,
                           const int* __restrict__ dst, const float* __restrict__ mx,
                           const float* __restrict__ denom, float* __restrict__ agg) {
    int tid = blockIdx.x * blockDim.x + threadIdx.x;
    if (tid >= NEDGES * 16) return;
    int e  = tid >> 4;
    int d4 = (tid & 15) * 4;
    int s = src[e], t = dst[e];
    float4 xv = *(const float4*)(x + (long)s * DFEAT + d4);
    float4 mv = *(const float4*)(mx + (long)t * DFEAT + d4);
    float4 dv = *(const float4*)(denom + (long)t * DFEAT + d4);
    float m0 = fmaxf(xv.x, 0.f) + GEPS;
    float m1 = fmaxf(xv.y, 0.f) + GEPS;
    float m2 = fmaxf(xv.z, 0.f) + GEPS;
    float m3 = fmaxf(xv.w, 0.f) + GEPS;
    float a0 = m0 * __expf(m0 - mv.x) / (dv.x + 1e-16f);
    float a1 = m1 * __expf(m1 - mv.y) / (dv.y + 1e-16f);
    float a2 = m2 * __expf(m2 - mv.z) / (dv.z + 1e-16f);
    float a3 = m3 * __expf(m3 - mv.w) / (dv.w + 1e-16f);
    float* p = agg + (long)t * DFEAT + d4;
    atomicAdd(p + 0, a0);
    atomicAdd(p + 1, a1);
    atomicAdd(p + 2, a2);
    atomicAdd(p + 3, a3);
}

// ------------------------- GEMM1: h = (agg + x) @ W1 + b1   [N,64]x[64,128]
// One wave per 16x16 output tile; residual fused into the A-fragment load.
// A frag (16x4 f32, 2 VGPRs): lanes 0-15 hold K=k0,k0+1; lanes 16-31 K=k0+2,k0+3.
__global__ void k_gemm1(const float* __restrict__ agg, const float* __restrict__ x,
                        const float* __restrict__ W1, const float* __restrict__ b1,
                        float* __restrict__ h) {
    const int lane = threadIdx.x;          // 0..31
    const int col0 = threadIdx.y * 16;     // 8 waves -> 8 col tiles of 128
    const int row0 = blockIdx.x * 16;
    const int half = lane >> 4;
    const int m    = lane & 15;
    const float* arow = agg + (long)(row0 + m) * DFEAT;
    const float* xrow = x   + (long)(row0 + m) * DFEAT;
    v8f acc = {};
#pragma unroll
    for (int k0 = 0; k0 < DFEAT; k0 += 4) {
        int k = k0 + 2 * half;
        v2f a, b;
        a.x = arow[k]     + xrow[k];
        a.y = arow[k + 1] + xrow[k + 1];
        b.x = W1[(long)k       * D2 + col0 + m];
        b.y = W1[(long)(k + 1) * D2 + col0 + m];
        acc = __builtin_amdgcn_wmma_f32_16x16x4_f32(false, a, false, b,
                                                    (short)0, acc, false, false);
    }
    float bias = b1[col0 + m];
#pragma unroll
    for (int r = 0; r < 8; ++r) {
        int mm = r + 8 * half;             // C layout: VGPR r -> M=r (lo half), r+8 (hi)
        h[(long)(row0 + mm) * D2 + col0 + m] = acc[r] + bias;
    }
}

// ------------------------------------- BN batch stats: column sum / sum-sq
__global__ void k_bnstats(const float* __restrict__ h, float* __restrict__ bnsum,
                          float* __restrict__ bnsq) {
    int c = threadIdx.x;                   // 0..127
    float s = 0.f, q = 0.f;
    for (int r = blockIdx.x; r < NNODES; r += gridDim.x) {
        float v = h[(long)r * D2 + c];
        s += v;
        q += v * v;
    }
    atomicAdd(&bnsum[c], s);
    atomicAdd(&bnsq[c], q);
}

__global__ void k_bnfin(const float* __restrict__ bnsum, const float* __restrict__ bnsq,
                        const float* __restrict__ gamma, const float* __restrict__ beta,
                        float* __restrict__ acoef, float* __restrict__ bcoef) {
    int c = threadIdx.x;                   // 128 threads
    float mean = bnsum[c] * (1.0f / NNODES);
    float var  = bnsq[c]  * (1.0f / NNODES) - mean * mean;
    float rstd = rsqrtf(var + 1e-5f);
    float a = gamma[c] * rstd;
    acoef[c] = a;
    bcoef[c] = beta[c] - mean * a;
}

// --------------- GEMM2: g = relu(h*acoef+bcoef) @ W2 + b2   [N,128]x[128,64]
__global__ void k_gemm2(const float* __restrict__ h, const float* __restrict__ acoef,
                        const float* __restrict__ bcoef, const float* __restrict__ W2,
                        const float* __restrict__ b2, float* __restrict__ g) {
    const int lane = threadIdx.x;
    const int col0 = threadIdx.y * 16;     // 4 waves -> 4 col tiles of 64
    const int row0 = blockIdx.x * 16;
    const int half = lane >> 4;
    const int m    = lane & 15;
    const float* hrow = h + (long)(row0 + m) * D2;
    v8f acc = {};
#pragma unroll
    for (int k0 = 0; k0 < D2; k0 += 4) {
        int k = k0 + 2 * half;
        v2f a, b;
        a.x = fmaxf(hrow[k]     * acoef[k]     + bcoef[k],     0.f);
        a.y = fmaxf(hrow[k + 1] * acoef[k + 1] + bcoef[k + 1], 0.f);
        b.x = W2[(long)k       * DFEAT + col0 + m];
        b.y = W2[(long)(k + 1) * DFEAT + col0 + m];
        acc = __builtin_amdgcn_wmma_f32_16x16x4_f32(false, a, false, b,
                                                    (short)0, acc, false, false);
    }
    float bias = b2[col0 + m];
#pragma unroll
    for (int r = 0; r < 8; ++r) {
        int mm = r + 8 * half;
        g[(long)(row0 + mm) * DFEAT + col0 + m] = acc[r] + bias;
    }
}

// --------------------------- LayerNorm(64) -> ReLU; one wave32 row, 2 cols/lane
__global__ void k_layernorm(const float* __restrict__ g, const float* __restrict__ lng,
                            const float* __restrict__ lnb, float* __restrict__ out) {
    int lane = threadIdx.x;                               // 0..31
    int row  = blockIdx.x * blockDim.y + threadIdx.y;     // 8 rows per block
    if (row >= NNODES) return;
    const float* grow = g + (long)row * DFEAT;
    float2 v = *(const float2*)(grow + lane * 2);
    float s = v.x + v.y;
    float q = v.x * v.x + v.y * v.y;
#pragma unroll
    for (int o = 16; o > 0; o >>= 1) {                    // wave32 tree reduce
        s += __shfl_xor(s, o, 32);
        q += __shfl_xor(q, o, 32);
    }
    float mu   = s * (1.0f / DFEAT);
    float var  = q * (1.0f / DFEAT) - mu * mu;
    float rstd = rsqrtf(var + 1e-5f);
    int c = lane * 2;
    float2 y;
    y.x = fmaxf((v.x - mu) * rstd * lng[c]     + lnb[c],     0.f);
    y.y = fmaxf((v.y - mu) * rstd * lng[c + 1] + lnb[c + 1], 0.f);
    *(float2*)(out + (long)row * DFEAT + c) = y;
}

// ----------------------------------------------------------------- launcher
extern "C" void kernel_launch(void* const* d_in, const int* in_sizes, int n_in,
                              void* d_out, int out_size, void* d_ws, size_t ws_size,
                              hipStream_t stream) {
    const float* x        = (const float*)d_in[0];
    const int*   eidx     = (const int*)d_in[1];
    const float* W1       = (const float*)d_in[2];
    const float* b1       = (const float*)d_in[3];
    const float* bn_gamma = (const float*)d_in[4];
    const float* bn_beta  = (const float*)d_in[5];
    const float* W2       = (const float*)d_in[6];
    const float* b2       = (const float*)d_in[7];
    const float* ln_gamma = (const float*)d_in[8];
    const float* ln_beta  = (const float*)d_in[9];
    float* out = (float*)d_out;

    const int* src = eidx;            // edge_index[0]
    const int* dst = eidx + NEDGES;   // edge_index[1]

    // workspace layout (floats)
    float* ws    = (float*)d_ws;
    float* mx    = ws;                               // N*64
    float* denom = mx    + (long)NNODES * DFEAT;     // N*64
    float* agg   = denom + (long)NNODES * DFEAT;     // N*64
    float* h     = agg   + (long)NNODES * DFEAT;     // N*128
    float* stats = h     + (long)NNODES * D2;        // 512: sum|sq|acoef|bcoef
    float* bnsum = stats;
    float* bnsq  = stats + 128;
    float* acoef = stats + 256;
    float* bcoef = stats + 384;
    float* g     = stats + 512;                      // N*64

    // 0) zero accumulators (every call: graph replays must be deterministic)
    k_zero<<<2048, 256, 0, stream>>>(mx, (long)3 * NNODES * DFEAT);
    k_zero<<<1, 256, 0, stream>>>(stats, 256);

    // 1..3) edge passes
    const int ethreads = NEDGES * 16;
    const int eblocks  = (ethreads + 255) / 256;
    k_edge_max   <<<eblocks, 256, 0, stream>>>(x, src, dst, mx);
    k_edge_expsum<<<eblocks, 256, 0, stream>>>(x, src, dst, mx, denom);
    k_edge_agg   <<<eblocks, 256, 0, stream>>>(x, src, dst, mx, denom, agg);

    // 4) GEMM1 (+fused residual)
    k_gemm1<<<NNODES / 16, dim3(32, 8), 0, stream>>>(agg, x, W1, b1, h);

    // 5..6) BatchNorm batch statistics
    k_bnstats<<<1024, 128, 0, stream>>>(h, bnsum, bnsq);
    k_bnfin<<<1, 128, 0, stream>>>(bnsum, bnsq, bn_gamma, bn_beta, acoef, bcoef);

    // 7) GEMM2 (+fused BN affine + ReLU on A)
    k_gemm2<<<NNODES / 16, dim3(32, 4), 0, stream>>>(h, acoef, bcoef, W2, b2, g);

    // 8) LayerNorm + ReLU
    k_layernorm<<<NNODES / 8, dim3(32, 8), 0, stream>>>(g, ln_gamma, ln_beta, out);
}